// modeler_46394236731752
// MI455X (gfx1250) — compile-verified
//
#include <hip/hip_runtime.h>

typedef __attribute__((ext_vector_type(2))) float v2f;
typedef __attribute__((ext_vector_type(8))) float v8f;

namespace {
constexpr int FT   = 256;   // input feature dim
constexpr int OUT  = 256;   // output feature dim
constexpr int NREL = 4;     // relations
constexpr int DEG  = 16;    // padded neighbors
constexpr int BM   = 16;    // batch rows per block (one WMMA M-tile)
constexpr int LDA  = 260;   // padded LDS stride (260%64==4 -> conflict-free A reads)
}

__global__ __launch_bounds__(256, 2) void rgcn_fused_wmma(
    const float* __restrict__ features,    // [100000, 256]
    const int*   __restrict__ nbr_idx,     // [8192, 4, 16]
    const float* __restrict__ W,           // [4, 256, 256]  (f-major, o-minor)
    const float* __restrict__ bias,        // [4, 256]
    const float* __restrict__ prelu_a,     // [1]
    float*       __restrict__ out)         // [8192, 256]
{
    __shared__ float vA[BM * LDA];         // v_in tile, 16 x 256 (padded)
    __shared__ int   idxs[BM * DEG];       // neighbor indices for current relation

    const int tid  = threadIdx.x;
    const int lane = tid & 31;
    const int wave = tid >> 5;             // 0..7
    const int row0 = blockIdx.x * BM;

    const float alpha = prelu_a[0];

    // ---- WMMA f32 16x16x4 lane mapping ----
    const int half = lane >> 4;            // 0: lanes 0-15, 1: lanes 16-31
    const int lm   = lane & 15;            // M (A/C) or N (B/C) index in tile
    const int koff = half * 2;             // K sub-offset within the 4-wide step

    const int ncol0 = wave * 32 + lm;      // this wave covers cols [wave*32, wave*32+32)
    const int ncol1 = ncol0 + 16;

    // ---- gather mapping: thread owns a float4 column slice of 4 rows ----
    const int col4 = (tid & 63) * 4;       // column (float index) of the float4
    const int rg   = tid >> 6;             // row group 0..3 -> rows rg, rg+4, rg+8, rg+12

    float outAcc0[8], outAcc1[8];
    #pragma unroll
    for (int v = 0; v < 8; ++v) { outAcc0[v] = 0.f; outAcc1[v] = 0.f; }

    for (int r = 0; r < NREL; ++r) {
        // ---------- stage neighbor indices for this (tile, relation) ----------
        {
            const int m = tid >> 4, d = tid & 15;
            idxs[tid] = nbr_idx[((size_t)(row0 + m) * NREL + r) * DEG + d];
        }
        __syncthreads();

        // ---------- gather + mean -> LDS v_in tile ----------
        float4 acc[4];
        #pragma unroll
        for (int j = 0; j < 4; ++j) acc[j] = make_float4(0.f, 0.f, 0.f, 0.f);

        #pragma unroll
        for (int j = 0; j < 4; ++j) {
            const int m = rg + 4 * j;
            #pragma unroll
            for (int d = 0; d < DEG; ++d) {
                const int nbr = idxs[m * DEG + d];
                const float4 f = *(const float4*)(features + (size_t)nbr * FT + col4);
                acc[j].x += f.x; acc[j].y += f.y; acc[j].z += f.z; acc[j].w += f.w;
            }
        }
        constexpr float inv_deg = 1.0f / (float)DEG;
        #pragma unroll
        for (int j = 0; j < 4; ++j) {
            const int m = rg + 4 * j;
            float4* dst = (float4*)&vA[m * LDA + col4];
            *dst = make_float4(acc[j].x * inv_deg, acc[j].y * inv_deg,
                               acc[j].z * inv_deg, acc[j].w * inv_deg);
        }
        __syncthreads();

        // ---------- per-relation GEMM: [16 x 256] x [256 x 256] via WMMA ----------
        v8f c0 = {};
        v8f c1 = {};
        const float* Wr = W + (size_t)r * FT * OUT;

        #pragma unroll 4
        for (int k = 0; k < FT; k += 4) {
            // A fragment: A[lm][k+koff .. k+koff+1] from LDS (b64, conflict-free)
            v2f a = *(const v2f*)&vA[lm * LDA + k + koff];
            // B fragments: W[k+koff][n], W[k+koff+1][n] (L2-resident)
            const float* wk = Wr + (size_t)(k + koff) * OUT;
            v2f b0; b0.x = wk[ncol0]; b0.y = wk[OUT + ncol0];
            v2f b1; b1.x = wk[ncol1]; b1.y = wk[OUT + ncol1];
            c0 = __builtin_amdgcn_wmma_f32_16x16x4_f32(
                     false, a, false, b0, (short)0, c0, false, false);
            c1 = __builtin_amdgcn_wmma_f32_16x16x4_f32(
                     false, a, false, b1, (short)0, c1, false, false);
        }

        // ---------- bias + PReLU, accumulate relation mean ----------
        const float bb0 = bias[r * OUT + ncol0];
        const float bb1 = bias[r * OUT + ncol1];
        #pragma unroll
        for (int v = 0; v < 8; ++v) {
            float h0 = c0[v] + bb0;
            float h1 = c1[v] + bb1;
            h0 = (h0 >= 0.f) ? h0 : alpha * h0;
            h1 = (h1 >= 0.f) ? h1 : alpha * h1;
            outAcc0[v] += h0;
            outAcc1[v] += h1;
        }
        __syncthreads();   // vA/idxs fully consumed before next relation overwrites
    }

    // ---------- write output: mean over relations ----------
    const int mbase = half * 8;            // C layout: lanes 16-31 hold M = v+8
    #pragma unroll
    for (int v = 0; v < 8; ++v) {
        const size_t row = (size_t)(row0 + v + mbase);
        out[row * OUT + ncol0] = outAcc0[v] * 0.25f;
        out[row * OUT + ncol1] = outAcc1[v] * 0.25f;
    }
}

extern "C" void kernel_launch(void* const* d_in, const int* in_sizes, int n_in,
                              void* d_out, int out_size, void* d_ws, size_t ws_size,
                              hipStream_t stream) {
    (void)in_sizes; (void)n_in; (void)d_ws; (void)ws_size; (void)out_size;
    const float* features = (const float*)d_in[0];
    const int*   nbr_idx  = (const int*)d_in[1];
    const float* W        = (const float*)d_in[2];
    const float* bias     = (const float*)d_in[3];
    const float* prelu_a  = (const float*)d_in[4];
    // d_in[5] (node_list) and d_in[6] (k) are unused by the reference math.
    float* out = (float*)d_out;

    const int BATCH = 8192;
    dim3 grid(BATCH / BM);   // 512 blocks, each owns a 16-row batch tile
    dim3 block(256);         // 8 waves: each wave owns 32 of the 256 output cols
    rgcn_fused_wmma<<<grid, block, 0, stream>>>(features, nbr_idx, W, bias, prelu_a, out);
}